// Decoder_6502580486208
// MI455X (gfx1250) — compile-verified
//
#include <hip/hip_runtime.h>
#include <math.h>

// ---------------------------------------------------------------------------
// LSTM decoder for MI455X (gfx1250), FP32 WMMA path (V_WMMA_F32_16X16X4_F32).
// Dims fixed by the reference: B=1024, Z=128, X=Y=64, H=512, T=128.
// ---------------------------------------------------------------------------

typedef __attribute__((ext_vector_type(2))) float v2f;
typedef __attribute__((ext_vector_type(8))) float v8f;

#define BATCH 1024
#define HDIM  512
#define XDIM  64
#define ZDIM  128
#define TSTEPS 128

__device__ __forceinline__ v8f wmma4f32(v2f a, v2f b, v8f c) {
    // D = A(16x4,f32) * B(4x16,f32) + C(16x16,f32)
    return __builtin_amdgcn_wmma_f32_16x16x4_f32(
        /*neg_a=*/false, a, /*neg_b=*/false, b,
        /*c_mod=*/(short)0, c, /*reuse_a=*/false, /*reuse_b=*/false);
}

__device__ __forceinline__ float sigm_f(float x) {
    return 1.0f / (1.0f + __expf(-x));
}
__device__ __forceinline__ float tanh_f(float x) {
    float e = __expf(-2.0f * fabsf(x));
    float r = (1.0f - e) / (1.0f + e);
    return x < 0.0f ? -r : r;
}

// ---------------------------------------------------------------------------
// y_prev0 = x[:, 31, :]
// ---------------------------------------------------------------------------
__global__ __launch_bounds__(256) void copy_y0_kernel(const float* __restrict__ x,
                                                      float* __restrict__ y_state) {
    int i = blockIdx.x * 256 + threadIdx.x;        // 0 .. 65535
    int m = i >> 6, n = i & 63;
    y_state[i] = x[m * (32 * XDIM) + 31 * XDIM + n];
}

// ---------------------------------------------------------------------------
// h0 = z @ W_proj^T + b_proj ; c0 = h0.   M=1024, N=512, K=128.
// Wave: 16 rows x 64 cols (4 N-subtiles). Block: 4 waves (64 rows). Grid (16,8).
// ---------------------------------------------------------------------------
__global__ __launch_bounds__(128) void init_kernel(const float* __restrict__ z,
                                                   const float* __restrict__ W_proj,
                                                   const float* __restrict__ b_proj,
                                                   float* __restrict__ h0,
                                                   float* __restrict__ c0) {
    const int lane = threadIdx.x & 31;
    const int wave = threadIdx.x >> 5;
    const int row  = lane & 15;      // M (for A) / N (for B) sub-index
    const int kh   = lane >> 4;      // K half-select
    const int m0   = blockIdx.x * 64 + wave * 16;
    const int nb   = blockIdx.y * 64;

    v8f acc[4];
    #pragma unroll
    for (int nt = 0; nt < 4; ++nt) {
        float bs = b_proj[nb + nt * 16 + row];
        #pragma unroll
        for (int r = 0; r < 8; ++r) acc[nt][r] = bs;
    }

    const float* Ar = z + (m0 + row) * ZDIM;
    #pragma unroll 4
    for (int k = 0; k < ZDIM; k += 4) {
        int ka = k + 2 * kh;
        v2f a = *(const v2f*)(Ar + ka);
        #pragma unroll
        for (int nt = 0; nt < 4; ++nt) {
            v2f b = *(const v2f*)(W_proj + (nb + nt * 16 + row) * ZDIM + ka);
            acc[nt] = wmma4f32(a, b, acc[nt]);
        }
    }

    #pragma unroll
    for (int nt = 0; nt < 4; ++nt) {
        #pragma unroll
        for (int r = 0; r < 8; ++r) {
            int m = m0 + r + 8 * kh;
            int idx = m * HDIM + nb + nt * 16 + row;
            float v = acc[nt][r];
            h0[idx] = v;
            c0[idx] = v;
        }
    }
}

// ---------------------------------------------------------------------------
// One LSTM step, fused: gates GEMM (K = 64 + 512) + pointwise LSTM update.
// Each wave owns 32 batch rows x 16 h-columns, computing the matching i/f/g/o
// tiles (N offsets 0/512/1024/1536 in the 2048-wide gate matrix), so the
// nonlinearity runs entirely in accumulator registers -- the 8MB gates matrix
// never hits memory. Block: 4 waves = 32 rows x 64 h-cols. Grid (32,8).
// h is double-buffered (h_prev read by all blocks, h_next written).
// ---------------------------------------------------------------------------
__global__ __launch_bounds__(128) void lstm_step_kernel(const float* __restrict__ y_prev,  // 1024x64
                                                        const float* __restrict__ h_prev,  // 1024x512
                                                        const float* __restrict__ W_ih,    // 2048x64
                                                        const float* __restrict__ W_hh,    // 2048x512
                                                        const float* __restrict__ b_ih,
                                                        const float* __restrict__ b_hh,
                                                        float* __restrict__ c_state,       // 1024x512
                                                        float* __restrict__ h_next) {      // 1024x512
    const int lane = threadIdx.x & 31;
    const int wave = threadIdx.x >> 5;
    const int row  = lane & 15;
    const int kh   = lane >> 4;
    const int m0   = blockIdx.x * 32;
    const int n0   = (blockIdx.y * 4 + wave) * 16;   // h-column group

    v8f acc[4][2];   // [gate i/f/g/o][m-subtile]
    #pragma unroll
    for (int g = 0; g < 4; ++g) {
        int gn = g * HDIM + n0 + row;
        float bs = b_ih[gn] + b_hh[gn];
        #pragma unroll
        for (int mt = 0; mt < 2; ++mt)
            #pragma unroll
            for (int r = 0; r < 8; ++r) acc[g][mt][r] = bs;
    }

    // --- y_prev @ W_ih^T, K = 64 ---
    {
        const float* A0 = y_prev + (m0 + row) * XDIM;
        const float* A1 = y_prev + (m0 + 16 + row) * XDIM;
        #pragma unroll 4
        for (int k = 0; k < XDIM; k += 4) {
            int ka = k + 2 * kh;
            v2f a0 = *(const v2f*)(A0 + ka);
            v2f a1 = *(const v2f*)(A1 + ka);
            #pragma unroll
            for (int g = 0; g < 4; ++g) {
                v2f b = *(const v2f*)(W_ih + (g * HDIM + n0 + row) * XDIM + ka);
                acc[g][0] = wmma4f32(a0, b, acc[g][0]);
                acc[g][1] = wmma4f32(a1, b, acc[g][1]);
            }
        }
    }
    // --- h_prev @ W_hh^T, K = 512 ---
    {
        const float* A0 = h_prev + (m0 + row) * HDIM;
        const float* A1 = h_prev + (m0 + 16 + row) * HDIM;
        #pragma unroll 4
        for (int k = 0; k < HDIM; k += 4) {
            int ka = k + 2 * kh;
            v2f a0 = *(const v2f*)(A0 + ka);
            v2f a1 = *(const v2f*)(A1 + ka);
            #pragma unroll
            for (int g = 0; g < 4; ++g) {
                v2f b = *(const v2f*)(W_hh + (g * HDIM + n0 + row) * HDIM + ka);
                acc[g][0] = wmma4f32(a0, b, acc[g][0]);
                acc[g][1] = wmma4f32(a1, b, acc[g][1]);
            }
        }
    }

    // --- pointwise LSTM update in registers; accumulator element (r, lane):
    //     m = m_base + r + 8*(lane>>4), n = n0 + (lane&15)  (ISA 7.12.2) ---
    #pragma unroll
    for (int mt = 0; mt < 2; ++mt) {
        #pragma unroll
        for (int r = 0; r < 8; ++r) {
            int m   = m0 + mt * 16 + r + 8 * kh;
            int idx = m * HDIM + n0 + row;
            float cv = c_state[idx];
            float iv = sigm_f(acc[0][mt][r]);
            float fv = sigm_f(acc[1][mt][r]);
            float gv = tanh_f(acc[2][mt][r]);
            float ov = sigm_f(acc[3][mt][r]);
            float cn = fv * cv + iv * gv;
            c_state[idx] = cn;
            h_next[idx]  = ov * tanh_f(cn);
        }
    }
}

// ---------------------------------------------------------------------------
// y_cur = y_prev + h @ W_out^T + b_out ; write to out[:, t, :] and y_state.
// M=1024, N=64, K=512. Wave: 16 rows x 64 cols. Grid: 16 blocks x 4 waves.
// ---------------------------------------------------------------------------
__global__ __launch_bounds__(128) void proj_kernel(const float* __restrict__ h,      // 1024x512
                                                   const float* __restrict__ W_out,  // 64x512
                                                   const float* __restrict__ b_out,  // 64
                                                   float* __restrict__ y_state,      // 1024x64
                                                   float* __restrict__ out,          // 1024x128x64
                                                   int t) {
    const int lane = threadIdx.x & 31;
    const int wave = threadIdx.x >> 5;
    const int row  = lane & 15;
    const int kh   = lane >> 4;
    const int m0   = (blockIdx.x * 4 + wave) * 16;

    v8f acc[4];
    #pragma unroll
    for (int nt = 0; nt < 4; ++nt) {
        float bs = b_out[nt * 16 + row];
        #pragma unroll
        for (int r = 0; r < 8; ++r) acc[nt][r] = bs;
    }

    const float* Ar = h + (m0 + row) * HDIM;
    #pragma unroll 4
    for (int k = 0; k < HDIM; k += 4) {
        int ka = k + 2 * kh;
        v2f a = *(const v2f*)(Ar + ka);
        #pragma unroll
        for (int nt = 0; nt < 4; ++nt) {
            v2f b = *(const v2f*)(W_out + (nt * 16 + row) * HDIM + ka);
            acc[nt] = wmma4f32(a, b, acc[nt]);
        }
    }

    #pragma unroll
    for (int nt = 0; nt < 4; ++nt) {
        #pragma unroll
        for (int r = 0; r < 8; ++r) {
            int m = m0 + r + 8 * kh;
            int n = nt * 16 + row;
            float yv = y_state[m * XDIM + n] + acc[nt][r];
            y_state[m * XDIM + n] = yv;
            out[m * (TSTEPS * XDIM) + t * XDIM + n] = yv;
        }
    }
}

// ---------------------------------------------------------------------------
extern "C" void kernel_launch(void* const* d_in, const int* in_sizes, int n_in,
                              void* d_out, int out_size, void* d_ws, size_t ws_size,
                              hipStream_t stream) {
    const float* z      = (const float*)d_in[0];
    const float* x      = (const float*)d_in[1];
    const float* W_ih   = (const float*)d_in[2];
    const float* W_hh   = (const float*)d_in[3];
    const float* b_ih   = (const float*)d_in[4];
    const float* b_hh   = (const float*)d_in[5];
    const float* W_proj = (const float*)d_in[6];
    const float* b_proj = (const float*)d_in[7];
    const float* W_out  = (const float*)d_in[8];
    const float* b_out  = (const float*)d_in[9];
    float* out = (float*)d_out;

    // Workspace layout (floats): h buffers (x2), c, y_prev  -> ~6.5 MB total
    float* ws   = (float*)d_ws;
    float* hbuf0 = ws;
    float* hbuf1 = ws + (size_t)BATCH * HDIM;
    float* cbuf  = ws + (size_t)2 * BATCH * HDIM;
    float* ybuf  = ws + (size_t)3 * BATCH * HDIM;

    copy_y0_kernel<<<(BATCH * XDIM) / 256, 256, 0, stream>>>(x, ybuf);
    init_kernel<<<dim3(BATCH / 64, HDIM / 64), 128, 0, stream>>>(z, W_proj, b_proj, hbuf0, cbuf);

    float* hc = hbuf0;
    float* hn = hbuf1;
    for (int t = 0; t < TSTEPS; ++t) {
        lstm_step_kernel<<<dim3(BATCH / 32, HDIM / 64), 128, 0, stream>>>(
            ybuf, hc, W_ih, W_hh, b_ih, b_hh, cbuf, hn);
        proj_kernel<<<BATCH / 64, 128, 0, stream>>>(hn, W_out, b_out, ybuf, out, t);
        float* tmp = hc; hc = hn; hn = tmp;
    }
}